// EdgeEnhanceLoss_8607114461410
// MI455X (gfx1250) — compile-verified
//
#include <hip/hip_runtime.h>
#include <hip/hip_bf16.h>

// B=8, C=1, H=W=256
#define HDIM 256
#define WDIM 256
#define BDIM 8
#define NPIX (BDIM * HDIM * WDIM)   // 524288
#define BIGF 1e12f

typedef __attribute__((ext_vector_type(2))) float v2f;
typedef __attribute__((ext_vector_type(8))) float v8f;

// ---------------------------------------------------------------------------
// Pass 1 (vertical): g[b,i,k] = min_m  f(m,k) + (i-m)^2,  f = target>0 ? 1e12 : 0
// One block per (i,b); 256 threads across columns k -> fully coalesced row loads.
// ---------------------------------------------------------------------------
__global__ __launch_bounds__(WDIM)
void edt_vertical(const int* __restrict__ target, float* __restrict__ g) {
    const int k = threadIdx.x;        // column
    const int i = blockIdx.x;         // output row
    const int b = blockIdx.y;         // image
    const int* img = target + b * (HDIM * WDIM);

    float best = 3.4e38f;
    for (int m = 0; m < HDIM; ++m) {
        // prefetch 8 rows ahead (clamped) -> global_prefetch_b8
        int pm = m + 8; if (pm > HDIM - 1) pm = HDIM - 1;
        __builtin_prefetch(img + pm * WDIM + k, 0, 1);

        const int   t = img[m * WDIM + k];
        const float f = (t > 0) ? BIGF : 0.0f;
        const float d = (float)(i - m);
        best = fminf(best, f + d * d);
    }
    g[(b * HDIM + i) * WDIM + k] = best;
}

// ---------------------------------------------------------------------------
// Pass 2 (horizontal) + weighted BCE + per-row partial sum.
// d2[b,i,j] = min_k g[b,i,k] + (j-k)^2
// w = exp(-d2/18); loss = -w * (t*max(log p,-100) + (1-t)*max(log1p(-p),-100))
// One block per (i,b); g row staged in LDS; tree reduction -> partial[b*256+i].
// ---------------------------------------------------------------------------
__global__ __launch_bounds__(WDIM)
void edt_horiz_loss(const float* __restrict__ g,
                    const float* __restrict__ pred,
                    const int* __restrict__ target,
                    float* __restrict__ partial) {
    __shared__ float row[WDIM];
    __shared__ float red[WDIM];

    const int j = threadIdx.x;
    const int i = blockIdx.x;
    const int b = blockIdx.y;
    const int base = (b * HDIM + i) * WDIM;

    row[j] = g[base + j];
    __syncthreads();

    float best = 3.4e38f;
    for (int k = 0; k < WDIM; ++k) {
        const float d = (float)(j - k);
        best = fminf(best, row[k] + d * d);
    }

    const float w = expf(best * (-1.0f / 18.0f));          // 2*sigma^2 = 18
    const float p = pred[base + j];
    const float t = (float)target[base + j];
    const float logp   = fmaxf(logf(p),     -100.0f);
    const float log1mp = fmaxf(log1pf(-p),  -100.0f);
    const float loss = -w * (t * logp + (1.0f - t) * log1mp);

    red[j] = loss;
    __syncthreads();
    for (int s = WDIM / 2; s > 0; s >>= 1) {
        if (j < s) red[j] += red[j + s];
        __syncthreads();
    }
    if (j == 0) partial[b * HDIM + i] = red[0];
}

// ---------------------------------------------------------------------------
// Final reduction of 2048 partials with V_WMMA_F32_16X16X4_F32 (exact f32 FMA):
// A = 16x4 tile of data, B = ones(4x16), C accumulates. D[m][n] = row-sums of A,
// replicated across columns; total = sum over M of any column N.
// A layout (32-bit, 16x4): lane L<16 holds (M=L,K=0),(M=L,K=1); lane 16+L holds
// (M=L,K=2),(M=L,K=3)  => per-lane flat base = (lane&15)*4 + (lane>>4)*2.
// Single wave32, uniform control flow => EXEC all ones as WMMA requires.
// ---------------------------------------------------------------------------
__global__ __launch_bounds__(32)
void wmma_reduce(const float* __restrict__ partial, float* __restrict__ out,
                 int n, float scale) {
    __shared__ float buf[32];
    const int lane = threadIdx.x;
    const int base = (lane & 15) * 4 + (lane >> 4) * 2;

    v2f ones; ones.x = 1.0f; ones.y = 1.0f;
    v8f c = {};
    for (int t0 = 0; t0 < n; t0 += 64) {
        const float* p = partial + t0 + base;
        v2f a; a.x = p[0]; a.y = p[1];
        c = __builtin_amdgcn_wmma_f32_16x16x4_f32(
                /*neg_a=*/false, a, /*neg_b=*/false, ones,
                /*c_mod=*/(short)0, c, /*reuse_a=*/false, /*reuse_b=*/false);
    }
    // lane L<16 holds D[M=0..7][N=L] in c[0..7]; lane 16+L holds D[M=8..15][N=L]
    float s = c[0] + c[1] + c[2] + c[3] + c[4] + c[5] + c[6] + c[7];
    buf[lane] = s;
    __syncthreads();
    if (lane == 0) out[0] = (buf[0] + buf[16]) * scale;
}

extern "C" void kernel_launch(void* const* d_in, const int* in_sizes, int n_in,
                              void* d_out, int out_size, void* d_ws, size_t ws_size,
                              hipStream_t stream) {
    const float* pred   = (const float*)d_in[0];   // [8,1,256,256] f32
    const int*   target = (const int*)d_in[1];     // [8,1,256,256] i32
    float* out = (float*)d_out;                    // scalar

    float* g       = (float*)d_ws;                 // 8*256*256 f32 = 2 MB
    float* partial = g + NPIX;                     // 2048 f32

    dim3 grid(HDIM, BDIM);
    edt_vertical<<<grid, WDIM, 0, stream>>>(target, g);
    edt_horiz_loss<<<grid, WDIM, 0, stream>>>(g, pred, target, partial);
    wmma_reduce<<<1, 32, 0, stream>>>(partial, out, BDIM * HDIM,
                                      1.0f / (float)NPIX);
}